// LotkaVolterra_40080634807112
// MI455X (gfx1250) — compile-verified
//
#include <hip/hip_runtime.h>
#include <stdint.h>

#define LV_TPB   128   // threads per block (4 wave32 waves), 1 particle/thread
#define LV_CHUNK 8     // RK4 steps staged in LDS per TDM tile store

#if defined(__gfx1250__) && __has_builtin(__builtin_amdgcn_tensor_store_from_lds) && \
    __has_builtin(__builtin_amdgcn_s_wait_tensorcnt)
#define LV_USE_TDM 1
#else
#define LV_USE_TDM 0
#endif

typedef unsigned int v4u __attribute__((ext_vector_type(4)));
typedef int          v4i __attribute__((ext_vector_type(4)));
typedef int          v8i __attribute__((ext_vector_type(8)));

// f(x,y): dx = alpha*x - beta*x*y ; dy = delta*x*y - gamma*y
__device__ __forceinline__ float2 lv_f(float2 s, float al, float be, float de, float ga) {
    float xy = s.x * s.y;
    float2 r;
    r.x = __builtin_fmaf(al, s.x, -(be * xy));
    r.y = __builtin_fmaf(de, xy, -(ga * s.y));
    return r;
}

// One RK4 step with step size h, fully in registers (~16-deep FMA chain).
__device__ __forceinline__ float2 lv_step(float2 s, float h,
                                          float al, float be, float de, float ga) {
    float2 k1 = lv_f(s, al, be, de, ga);  k1.x *= h;  k1.y *= h;
    float2 s2 = { __builtin_fmaf(0.5f, k1.x, s.x), __builtin_fmaf(0.5f, k1.y, s.y) };
    float2 k2 = lv_f(s2, al, be, de, ga); k2.x *= h;  k2.y *= h;
    float2 s3 = { __builtin_fmaf(0.5f, k2.x, s.x), __builtin_fmaf(0.5f, k2.y, s.y) };
    float2 k3 = lv_f(s3, al, be, de, ga); k3.x *= h;  k3.y *= h;
    float2 s4 = { s.x + k3.x, s.y + k3.y };
    float2 k4 = lv_f(s4, al, be, de, ga); k4.x *= h;  k4.y *= h;
    s.x += (k1.x + 2.0f * (k2.x + k3.x) + k4.x) * (1.0f / 6.0f);
    s.y += (k1.y + 2.0f * (k2.y + k3.y) + k4.y) * (1.0f / 6.0f);
    return s;
}

__global__ __launch_bounds__(LV_TPB) void lv_rk4_kernel(
    const float2* __restrict__ x0, const float* __restrict__ ts,
    const float* __restrict__ pa, const float* __restrict__ pb,
    const float* __restrict__ pd, const float* __restrict__ pg,
    float2* __restrict__ out, int B, int T) {

    const int tid = threadIdx.x;
    const int gid = blockIdx.x * LV_TPB + tid;
    const float al = pa[0], be = pb[0], de = pd[0], ga = pg[0];

    float2 s = x0[gid];
    out[gid] = s;                      // X[0] = x0 (row 0)

    const int steps = T - 1;
    float tprev = ts[0];               // uniform -> scalar load

#if LV_USE_TDM
    __shared__ float2 buf[2][LV_CHUNK][LV_TPB];       // 16 KB, double-buffered
    // Low 32 bits of the flat address of an LDS object == byte offset in the
    // workgroup's LDS allocation (aperture places the tag in addr[63:32]).
    const uint32_t lds0 = (uint32_t)(uintptr_t)(&buf[0][0][0]);
    const uint32_t lds_half = (uint32_t)(LV_CHUNK * LV_TPB * sizeof(float2)); // 8 KB
    const uint32_t tile_d0  = LV_TPB * 2u;            // floats per tile row (1 KB)
    const uint64_t stride0  = (uint64_t)(2 * B);      // floats between time rows
    const uint64_t row_bytes = (uint64_t)B * sizeof(float2);

    // Global byte address of this block's slice of row 1.
    uint64_t gaddr = (uint64_t)(uintptr_t)(out + (uint64_t)B + (uint64_t)blockIdx.x * LV_TPB);

    int i = 0, c = 0;
    while (i < steps) {
        const int rows = (steps - i) < LV_CHUNK ? (steps - i) : LV_CHUNK;
        const int bsel = c & 1;

        // Protect buf[bsel]: TDM store of chunk c-2 (same buffer) must be done.
        if (tid < 32) __builtin_amdgcn_s_wait_tensorcnt(1);
        __syncthreads();

        if (rows == LV_CHUNK) {
            // Batch the uniform ts loads (one s_load clause, one KMcnt wait),
            // then a fully unrolled 8-step chain the scheduler can pipeline.
            float t[LV_CHUNK];
#pragma unroll
            for (int j = 0; j < LV_CHUNK; ++j) t[j] = ts[i + j + 1];
#pragma unroll
            for (int j = 0; j < LV_CHUNK; ++j) {
                float h = t[j] - tprev;
                tprev = t[j];
                s = lv_step(s, h, al, be, de, ga);
                buf[bsel][j][tid] = s;                 // ds_store_b64
            }
        } else {
            for (int j = 0; j < rows; ++j) {
                float tnext = ts[i + j + 1];
                float h = tnext - tprev;
                tprev = tnext;
                s = lv_step(s, h, al, be, de, ga);
                buf[bsel][j][tid] = s;
            }
        }
        __syncthreads();                               // LDS writes visible (DScnt drained)

        if (tid < 32) {                                // one wave issues the bulk DMA
            const uint32_t tile_d1 = (uint32_t)rows;

            // ---- D# group 0 (ISA 8.3): count=1, lds_addr, global_addr, type=2 ----
            v4u g0;
            g0[0] = 1u;
            g0[1] = lds0 + (uint32_t)bsel * lds_half;
            g0[2] = (uint32_t)(gaddr & 0xFFFFFFFFu);
            g0[3] = (uint32_t)((gaddr >> 32) & 0x01FFFFFFu) | (2u << 30);

            // ---- D# group 1 (ISA 8.4): data_size=4B, dims, tile dims, dim0 stride ----
            v8i g1;
            g1[0] = (int)(2u << 16);                                        // data_size = 4 bytes
            g1[1] = (int)((tile_d0 & 0xFFFFu) << 16);                       // tensor_dim0[15:0]
            g1[2] = (int)(((tile_d0 >> 16) & 0xFFFFu) |
                          ((tile_d1 & 0xFFFFu) << 16));                     // dim0[31:16] | dim1[15:0]
            g1[3] = (int)(((tile_d1 >> 16) & 0xFFFFu) |
                          ((tile_d0 & 0xFFFFu) << 16));                     // dim1[31:16] | tile_dim0
            g1[4] = (int)(tile_d1 & 0xFFFFu);                               // tile_dim1 | tile_dim2=0
            g1[5] = (int)(uint32_t)(stride0 & 0xFFFFFFFFu);                 // dim0_stride[31:0]
            g1[6] = (int)(uint32_t)((stride0 >> 32) & 0xFFFFu);             // dim0_stride[47:32]
            g1[7] = 0;

            v4i g2 = {0, 0, 0, 0};   // 2-D tile: tile_dim2/3 unused
            v4i g3 = {0, 0, 0, 0};
#if __clang_major__ >= 23
            v8i g4 = {0, 0, 0, 0, 0, 0, 0, 0};
            __builtin_amdgcn_tensor_store_from_lds(g0, g1, g2, g3, g4, 0);
#else
            __builtin_amdgcn_tensor_store_from_lds(g0, g1, g2, g3, 0);
#endif
        }
        gaddr += (uint64_t)rows * row_bytes;
        i += rows;
        ++c;
    }
    // s_endpgm performs an implicit wait-idle, draining outstanding TENSORcnt.
#else
    // Fallback: plain coalesced global_store_b64 per step (fire-and-forget).
    for (int i = 0; i < steps; ++i) {
        float tnext = ts[i + 1];
        float h = tnext - tprev;
        tprev = tnext;
        s = lv_step(s, h, al, be, de, ga);
        out[(size_t)(i + 1) * (size_t)B + gid] = s;
    }
#endif
}

extern "C" void kernel_launch(void* const* d_in, const int* in_sizes, int n_in,
                              void* d_out, int out_size, void* d_ws, size_t ws_size,
                              hipStream_t stream) {
    const float2* x0 = (const float2*)d_in[0];
    const float*  ts = (const float*)d_in[1];
    const float*  pa = (const float*)d_in[2];
    const float*  pb = (const float*)d_in[3];
    const float*  pd = (const float*)d_in[4];
    const float*  pg = (const float*)d_in[5];
    float2* out = (float2*)d_out;

    const int B = in_sizes[0] / 2;   // 16384 particles
    const int T = in_sizes[1];       // 1001 time points

    dim3 block(LV_TPB);
    dim3 grid(B / LV_TPB);           // 16384 -> 128 blocks of 4 wave32 waves
    lv_rk4_kernel<<<grid, block, 0, stream>>>(x0, ts, pa, pb, pd, pg, out, B, T);
}